// PaddingBottleneck_70334384439458
// MI455X (gfx1250) — compile-verified
//
#include <hip/hip_runtime.h>

// CDNA5 / gfx1250, wave32.
typedef float v2f __attribute__((ext_vector_type(2)));
typedef float v4f __attribute__((ext_vector_type(4)));
typedef float v8f __attribute__((ext_vector_type(8)));

#define EDIM 1024
#define WAVES_PER_BLOCK 8   // 256 threads = 8 wave32 waves; one row per wave

// Broadcast lane 0's value to all lanes via SALU readfirstlane (lane 0 always active here).
__device__ __forceinline__ float bcast_lane0(float v) {
    return __builtin_bit_cast(float,
        __builtin_amdgcn_readfirstlane(__builtin_bit_cast(int, v)));
}

// SWAPX16: ds_swizzle group-of-32, xor_mask=0x10, or=0, and_mask=0x1f -> offset 0x401F.
__device__ __forceinline__ float swapx16(float v) {
    return __builtin_bit_cast(float,
        __builtin_amdgcn_ds_swizzle(__builtin_bit_cast(int, v), 0x401F));
}

__global__ __launch_bounds__(WAVES_PER_BLOCK * 32)
void padding_bottleneck_kernel(const float* __restrict__ x,
                               const float* __restrict__ padding_amount,
                               float* __restrict__ out,
                               int nrows)
{
    const int lane = threadIdx.x & 31;
    const int wave = threadIdx.x >> 5;
    const int row  = blockIdx.x * WAVES_PER_BLOCK + wave;
    if (row >= nrows) return;            // wave-uniform: EXEC stays all-1s for WMMA

    const v4f* __restrict__ rp = (const v4f*)(x   + (size_t)row * EDIM);
    v4f*       __restrict__ op = (v4f*)      (out + (size_t)row * EDIM);

    // ---- single streaming read of the row: 8 x b128 per lane, NT hint ----
    v4f d[8];
#pragma unroll
    for (int i = 0; i < 8; ++i)
        d[i] = __builtin_nontemporal_load(&rp[lane + 32 * i]);

    // per-lane partial abs-sum
    float partial = 0.0f;
#pragma unroll
    for (int i = 0; i < 8; ++i)
        partial += (fabsf(d[i].x) + fabsf(d[i].y)) + (fabsf(d[i].z) + fabsf(d[i].w));

    // ---- wave reduction via V_WMMA_F32_16X16X4_F32 against all-ones B ----
    // A layout (16x4 f32): lane L, VGPR0 holds A[L%16][2*(L/16)] -> partial in a.x.
    // With B = ones: D[m][n] = partial[m] + partial[m+16], identical across columns.
    v2f a; a.x = partial; a.y = 0.0f;
    v2f b; b.x = 1.0f;    b.y = 1.0f;
    v8f c = {};
    c = __builtin_amdgcn_wmma_f32_16x16x4_f32(
            /*neg_a=*/false, a, /*neg_b=*/false, b,
            /*c_mod=*/(short)0, c, /*reuse_a=*/false, /*reuse_b=*/false);

    // lanes 0-15 hold D[0..7][n], lanes 16-31 hold D[8..15][n]
    float s = ((c[0] + c[1]) + (c[2] + c[3])) + ((c[4] + c[5]) + (c[6] + c[7]));
    float total = s + swapx16(s);                     // full 32-lane sum, every lane

    // ---- scalar row math (identical in every lane) ----
    float padv        = bcast_lane0(d[0].x);          // element 0 of the row
    float signal_norm = total - fabsf(padv);          // exclude the pad element
    // softmax([pad, norm])[1] = sigmoid(norm - pad) = 1/(1 + 2^((pad-norm)*log2e))
    float t         = (padv - signal_norm) * 1.442695040888963f;
    float one_minus = __builtin_amdgcn_rcpf(1.0f + __builtin_amdgcn_exp2f(t));
    float pscale    = 1.0f + fabsf(padding_amount[0]);

    // ---- single streaming write of the row ----
#pragma unroll
    for (int i = 0; i < 8; ++i) {
        v4f o;
        o.x = d[i].x * one_minus;
        o.y = d[i].y * one_minus;
        o.z = d[i].z * one_minus;
        o.w = d[i].w * one_minus;
        if (i == 0 && lane == 0) o.x = padv * pscale; // pad slot gets pad*(1+|amt|)
        __builtin_nontemporal_store(o, &op[lane + 32 * i]);
    }
}

extern "C" void kernel_launch(void* const* d_in, const int* in_sizes, int n_in,
                              void* d_out, int out_size, void* d_ws, size_t ws_size,
                              hipStream_t stream) {
    const float* x       = (const float*)d_in[0];
    const float* pad_amt = (const float*)d_in[1];
    float*       out     = (float*)d_out;

    const int nrows  = in_sizes[0] / EDIM;                         // B*T = 65536
    const int blocks = (nrows + WAVES_PER_BLOCK - 1) / WAVES_PER_BLOCK;

    padding_bottleneck_kernel<<<blocks, WAVES_PER_BLOCK * 32, 0, stream>>>(
        x, pad_amt, out, nrows);
}